// GAT_87488483820067
// MI455X (gfx1250) — compile-verified
//
#include <hip/hip_runtime.h>
#include <math.h>

typedef float v2f __attribute__((ext_vector_type(2)));
typedef float v8f __attribute__((ext_vector_type(8)));

#define NEG_SLOPE 0.2f
#define EPSV 1e-16f

__device__ __forceinline__ float lrelu(float v) { return v > 0.0f ? v : NEG_SLOPE * v; }

__device__ __forceinline__ void atomic_add_f32(float* p, float v) {
  __hip_atomic_fetch_add(p, v, __ATOMIC_RELAXED, __HIP_MEMORY_SCOPE_AGENT);
}

// float atomic max via monotonic int mapping: signed max for >=0, unsigned min for <0.
__device__ __forceinline__ void atomic_max_f32(float* p, float v) {
  if (v >= 0.0f)
    __hip_atomic_fetch_max((int*)p, __float_as_int(v), __ATOMIC_RELAXED, __HIP_MEMORY_SCOPE_AGENT);
  else
    __hip_atomic_fetch_min((unsigned int*)p, __float_as_uint(v), __ATOMIC_RELAXED, __HIP_MEMORY_SCOPE_AGENT);
}

__global__ void fill_f32(float* __restrict__ p, float v, int n) {
  int i = blockIdx.x * blockDim.x + threadIdx.x;
  if (i < n) p[i] = v;
}

// C[M,Nc] = A[M,K] x B[K,Nc], all f32 row-major, exact f32 via V_WMMA_F32_16X16X4_F32.
// One wave (32 lanes) per 16x16 C tile. grid = (tilesN, tilesM), block = 32 (wave32).
// A frag (16x4, MxK): lanes 0-15 rows m, VGPR0/1 = K {k0,k0+1}; lanes 16-31 = K {k0+2,k0+3}.
// B frag (4x16, KxN): lane = column n; same K-half split across lane halves.
// C/D frag: VGPR v holds C[tm*16 + v + 8*(lane>>4)][tn*16 + (lane&15)].
__global__ void wmma_gemm_f32(const float* __restrict__ A, const float* __restrict__ B,
                              float* __restrict__ C, int M, int Nc, int K) {
  const int lane = threadIdx.x;
  const int tn = blockIdx.x, tm = blockIdx.y;
  const int half = lane >> 4;
  const int l16 = lane & 15;
  int row = tm * 16 + l16;
  int rA = row < M ? row : M - 1;          // clamp (M may not be multiple of 16)
  int col = tn * 16 + l16;
  int cB = col < Nc ? col : Nc - 1;        // clamp cols (Nc=40 -> 3 tiles)
  v8f acc = {};
  for (int k0 = 0; k0 < K; k0 += 4) {
    int ka = k0 + 2 * half;
    v2f a, b;
    const float* ap = A + (size_t)rA * K + ka;
    a.x = ap[0];
    a.y = ap[1];
    const float* bp = B + (size_t)ka * Nc + cB;
    b.x = bp[0];
    b.y = bp[Nc];
    acc = __builtin_amdgcn_wmma_f32_16x16x4_f32(false, a, false, b, (short)0, acc,
                                                false, false);
  }
  if (col < Nc) {
    int mbase = tm * 16 + half * 8;
    for (int v = 0; v < 8; ++v) {
      int m = mbase + v;
      if (m < M) C[(size_t)m * Nc + col] = acc[v];
    }
  }
}

// a_src[n,h] = h[n,h,:] . att_src[h,:] ; a_dst likewise.
template <int H, int C>
__global__ void att_proj(const float* __restrict__ h, const float* __restrict__ att_s,
                         const float* __restrict__ att_d, float* __restrict__ as,
                         float* __restrict__ ad, int N) {
  int t = blockIdx.x * blockDim.x + threadIdx.x;
  if (t >= N * H) return;
  int n = t / H, hd = t % H;
  const float* hp = h + (size_t)n * (H * C) + hd * C;
  float ss = 0.f, sd = 0.f;
#pragma unroll
  for (int c = 0; c < C; ++c) {
    ss += hp[c] * att_s[hd * C + c];
    sd += hp[c] * att_d[hd * C + c];
  }
  as[t] = ss;
  ad[t] = sd;
}

// Pass 1: segment max of leaky_relu(a_src[s]+a_dst[d]) into emax[d]. Edges e<E from
// edge_index, e>=E are the N self-loops.
template <int H>
__global__ void edge_max(const int* __restrict__ src, const int* __restrict__ dst, int E,
                         int N, const float* __restrict__ as, const float* __restrict__ ad,
                         float* __restrict__ emax) {
  int e = blockIdx.x * blockDim.x + threadIdx.x;
  if (e >= E + N) return;
  int s, d;
  if (e < E) { s = src[e]; d = dst[e]; } else { s = e - E; d = s; }
#pragma unroll
  for (int hd = 0; hd < H; ++hd) {
    float v = lrelu(as[s * H + hd] + ad[d * H + hd]);
    atomic_max_f32(&emax[d * H + hd], v);
  }
}

// Pass 2: w = exp(e - emax[d]); denom[d]+=w; agg[d,:] += w * h[s,:]  (unnormalized).
template <int H, int C>
__global__ void edge_accum(const int* __restrict__ src, const int* __restrict__ dst, int E,
                           int N, const float* __restrict__ as, const float* __restrict__ ad,
                           const float* __restrict__ emax, const float* __restrict__ h,
                           float* __restrict__ denom, float* __restrict__ agg) {
  int e = blockIdx.x * blockDim.x + threadIdx.x;
  if (e >= E + N) return;
  int s, d;
  if (e < E) { s = src[e]; d = dst[e]; } else { s = e - E; d = s; }
#pragma unroll
  for (int hd = 0; hd < H; ++hd) {
    float v = lrelu(as[s * H + hd] + ad[d * H + hd]);
    float w = expf(v - emax[d * H + hd]);
    atomic_add_f32(&denom[d * H + hd], w);
    const float* hs = h + (size_t)s * (H * C) + hd * C;
    float* ag = agg + (size_t)d * (H * C) + hd * C;
#pragma unroll
    for (int c = 0; c < C; ++c) atomic_add_f32(&ag[c], w * hs[c]);
  }
}

// Pass 3 (layer 1): normalize, add bias, ReLU. In place.
template <int H, int C>
__global__ void norm_bias_relu(float* __restrict__ agg, const float* __restrict__ denom,
                               const float* __restrict__ bias, int N) {
  int t = blockIdx.x * blockDim.x + threadIdx.x;
  if (t >= N * H * C) return;
  int n = t / (H * C);
  int i = t % (H * C);
  int hd = i / C;
  float v = agg[t] / (denom[n * H + hd] + EPSV) + bias[i];
  agg[t] = v > 0.f ? v : 0.f;
}

// Pass 3 (layer 2) + log_softmax over C2=40 classes.
__global__ void norm_logsoftmax(const float* __restrict__ agg, const float* __restrict__ denom,
                                const float* __restrict__ bias, float* __restrict__ out, int N) {
  const int C2 = 40;
  int n = blockIdx.x * blockDim.x + threadIdx.x;
  if (n >= N) return;
  float vals[C2];
  float dn = denom[n] + EPSV;
  float mx = -INFINITY;
#pragma unroll
  for (int c = 0; c < C2; ++c) {
    float v = agg[(size_t)n * C2 + c] / dn + bias[c];
    vals[c] = v;
    mx = fmaxf(mx, v);
  }
  float s = 0.f;
#pragma unroll
  for (int c = 0; c < C2; ++c) s += expf(vals[c] - mx);
  float l = mx + logf(s);
#pragma unroll
  for (int c = 0; c < C2; ++c) out[(size_t)n * C2 + c] = vals[c] - l;
}

static inline int cdiv(long long a, long long b) { return (int)((a + b - 1) / b); }

extern "C" void kernel_launch(void* const* d_in, const int* in_sizes, int n_in,
                              void* d_out, int out_size, void* d_ws, size_t ws_size,
                              hipStream_t stream) {
  const float* x        = (const float*)d_in[0];
  const int*   ei       = (const int*)d_in[1];
  const float* W1       = (const float*)d_in[2];
  const float* att_src1 = (const float*)d_in[3];
  const float* att_dst1 = (const float*)d_in[4];
  const float* b1       = (const float*)d_in[5];
  const float* W2       = (const float*)d_in[6];
  const float* att_src2 = (const float*)d_in[7];
  const float* att_dst2 = (const float*)d_in[8];
  const float* b2       = (const float*)d_in[9];

  const int FIN = 256, H1 = 2, C1 = 16, HC1 = H1 * C1, C2 = 40;
  const int N = in_sizes[0] / FIN;
  const int E = in_sizes[1] / 2;
  const int* src = ei;
  const int* dst = ei + E;
  const int ET = E + N;  // edges incl. self-loops

  // ---- workspace carve-up (all f32) ----
  float* p = (float*)d_ws;
  float* h1     = p; p += (size_t)N * HC1;   // [N,32] x@W1
  float* as1    = p; p += (size_t)N * H1;    // [N,2]
  float* ad1    = p; p += (size_t)N * H1;
  float* emax1  = p; p += (size_t)N * H1;
  float* den1   = p; p += (size_t)N * H1;
  float* agg1   = p; p += (size_t)N * HC1;   // -> becomes h2 after norm/relu
  float* g2     = p; p += (size_t)N * C2;    // [N,40] h2@W2
  float* as2    = p; p += (size_t)N;
  float* ad2    = p; p += (size_t)N;
  float* emax2  = p; p += (size_t)N;
  float* den2   = p; p += (size_t)N;
  float* agg2   = p; p += (size_t)N * C2;
  float* out    = (float*)d_out;

  const int TB = 256;
  const float NEG_INF = -INFINITY;

  // ---- init accumulators ----
  fill_f32<<<cdiv((size_t)N * H1, TB), TB, 0, stream>>>(emax1, NEG_INF, N * H1);
  fill_f32<<<cdiv((size_t)N * H1, TB), TB, 0, stream>>>(den1, 0.f, N * H1);
  fill_f32<<<cdiv((size_t)N * HC1, TB), TB, 0, stream>>>(agg1, 0.f, N * HC1);
  fill_f32<<<cdiv(N, TB), TB, 0, stream>>>(emax2, NEG_INF, N);
  fill_f32<<<cdiv(N, TB), TB, 0, stream>>>(den2, 0.f, N);
  fill_f32<<<cdiv((size_t)N * C2, TB), TB, 0, stream>>>(agg2, 0.f, N * C2);

  // ---- layer 1 ----
  {
    dim3 grid(HC1 / 16, cdiv(N, 16));  // 2 x 6250 tiles
    wmma_gemm_f32<<<grid, 32, 0, stream>>>(x, W1, h1, N, HC1, FIN);
  }
  att_proj<H1, C1><<<cdiv((size_t)N * H1, TB), TB, 0, stream>>>(h1, att_src1, att_dst1,
                                                                as1, ad1, N);
  edge_max<H1><<<cdiv(ET, TB), TB, 0, stream>>>(src, dst, E, N, as1, ad1, emax1);
  edge_accum<H1, C1><<<cdiv(ET, TB), TB, 0, stream>>>(src, dst, E, N, as1, ad1, emax1,
                                                      h1, den1, agg1);
  norm_bias_relu<H1, C1><<<cdiv((size_t)N * HC1, TB), TB, 0, stream>>>(agg1, den1, b1, N);
  // agg1 now holds h2 = relu(gat1 output), [N,32]

  // ---- layer 2 ----
  {
    dim3 grid(cdiv(C2, 16), cdiv(N, 16));  // 3 x 6250 tiles (cols 40..47 masked)
    wmma_gemm_f32<<<grid, 32, 0, stream>>>(agg1, W2, g2, N, C2, HC1);
  }
  att_proj<1, C2><<<cdiv(N, TB), TB, 0, stream>>>(g2, att_src2, att_dst2, as2, ad2, N);
  edge_max<1><<<cdiv(ET, TB), TB, 0, stream>>>(src, dst, E, N, as2, ad2, emax2);
  edge_accum<1, C2><<<cdiv(ET, TB), TB, 0, stream>>>(src, dst, E, N, as2, ad2, emax2,
                                                     g2, den2, agg2);
  norm_logsoftmax<<<cdiv(N, TB), TB, 0, stream>>>(agg2, den2, b2, out, N);
}